// MultiScaleDecoder_28252294873696
// MI455X (gfx1250) — compile-verified
//
#include <hip/hip_runtime.h>

typedef __attribute__((ext_vector_type(2))) float v2f;
typedef __attribute__((ext_vector_type(8))) float v8f;

#define BB   512
#define NTOK 4352   // 256 coarse + 4096 fine
#define NC   256
#define NF   4096
#define EE   32

// ---------------------------------------------------------------------------
// K1: token projections via V_WMMA_F32_16X16X4_F32.
// One wave = one 16-token tile. A (16x4 f32) = token slab, B (4x16) holds the
// 32-long weight vector in column N=0, 8 WMMAs cover K=32. Column 0 of the
// 16x16 f32 D lives in lanes 0 (M=0..7) and 16 (M=8..15).
// ---------------------------------------------------------------------------
__global__ __launch_bounds__(256) void proj_kernel(
    const float* __restrict__ tokens,
    const float* __restrict__ w_coarse, const float* __restrict__ b_coarse,
    const float* __restrict__ w_fine,   const float* __restrict__ b_fine,
    float* __restrict__ coarse_feats,   float* __restrict__ fine_feats)
{
    const int wave = (int)((blockIdx.x * blockDim.x + threadIdx.x) >> 5);
    const int lane = (int)(threadIdx.x & 31);
    const int tiles_per_batch = NTOK / 16;           // 272
    const int b    = wave / tiles_per_batch;
    const int t0   = (wave % tiles_per_batch) * 16;  // first token of tile
    const bool is_coarse = (t0 < NC);

    const float* w    = is_coarse ? w_coarse : w_fine;
    const float  bias = is_coarse ? b_coarse[0] : b_fine[0];

    const int half = lane >> 4;   // K sub-pair selector
    const int m    = lane & 15;   // A row / B,C column

    const float* arow = tokens + ((size_t)b * NTOK + (size_t)(t0 + m)) * EE;
    const float  wsel = (m == 0) ? 1.0f : 0.0f;  // only B column N=0 is non-zero

    v8f c = {};
#pragma unroll
    for (int j = 0; j < 8; ++j) {
        const int kb = 4 * j + 2 * half;
        v2f a = __builtin_nontemporal_load((const v2f*)(arow + kb));  // streamed once
        v2f bmat;
        bmat.x = wsel * w[kb];
        bmat.y = wsel * w[kb + 1];
        c = __builtin_amdgcn_wmma_f32_16x16x4_f32(
                /*neg_a=*/false, a, /*neg_b=*/false, bmat,
                /*c_mod=*/(short)0, c, /*reuse_a=*/false, /*reuse_b=*/false);
    }

    if (m == 0) {  // lanes 0 and 16 hold column N=0
        const int mbase = half * 8;
        float* dst = is_coarse ? (coarse_feats + (size_t)b * NC + t0)
                               : (fine_feats   + (size_t)b * NF + (t0 - NC));
#pragma unroll
        for (int r = 0; r < 8; ++r) dst[mbase + r] = c[r] + bias;
    }
}

// ---------------------------------------------------------------------------
// K2: per-batch mask decompaction. rank = cumsum(mask)-1; masked position i
// reads fine_feats[rank], unmasked -> 0. Block per batch, 16 elems/thread.
// ---------------------------------------------------------------------------
__global__ __launch_bounds__(256) void scatter_kernel(
    const int* __restrict__ mask, const float* __restrict__ fine_feats,
    float* __restrict__ fine_map)
{
    const int b = blockIdx.x;
    const int t = threadIdx.x;
    const int* mrow = mask       + (size_t)b * NF;
    const float* fr = fine_feats + (size_t)b * NF;
    float* orow     = fine_map   + (size_t)b * NF;

    __shared__ int sums[256];
    int loc[16];
    int s = 0;
    const int base = t * 16;
#pragma unroll
    for (int i = 0; i < 16; ++i) { loc[i] = mrow[base + i]; s += loc[i]; }
    sums[t] = s;
    __syncthreads();

    int run = 0;
    for (int i = 0; i < t; ++i) run += sums[i];  // tiny serial prefix (256 adds)

#pragma unroll
    for (int i = 0; i < 16; ++i) {
        float v = 0.0f;
        if (loc[i]) { v = fr[run]; ++run; }
        orow[base + i] = v;
    }
}

// ---------------------------------------------------------------------------
// K3: per-batch fused tail: bilinear 16->32 and 64->32 into zero-halo LDS
// tiles, conv1(2->2,3x3,SAME)+ReLU in LDS, conv2(2->1,3x3,SAME) to d_out.
// ---------------------------------------------------------------------------
__global__ __launch_bounds__(256) void decode_kernel(
    const float* __restrict__ coarse_feats,  // (B,256) as 16x16
    const float* __restrict__ fine_map,      // (B,4096) as 64x64
    const float* __restrict__ w_conv1, const float* __restrict__ b_conv1,
    const float* __restrict__ w_conv2, const float* __restrict__ b_conv2,
    float* __restrict__ out)                 // (B,1,32,32)
{
    const int b = blockIdx.x;
    const int t = threadIdx.x;

    __shared__ float fused[2][34][34];  // zero halo -> branch-free SAME conv
    __shared__ float h[2][34][34];
    __shared__ float wc1[36], wc2[18], bc1[2];
    __shared__ float bc2s;

    // zero both LDS tiles (covers halos)
    for (int i = t; i < 2 * 34 * 34; i += 256) {
        (&fused[0][0][0])[i] = 0.0f;
        (&h[0][0][0])[i]     = 0.0f;
    }
    if (t < 36) wc1[t] = w_conv1[t];
    if (t < 18) wc2[t] = w_conv2[t];
    if (t < 2)  bc1[t] = b_conv1[t];
    if (t == 0) bc2s   = b_conv2[0];
    __syncthreads();

    const float* cf = coarse_feats + (size_t)b * NC;
    const float* fm = fine_map     + (size_t)b * NF;

    // bilinear upsample 16x16 -> 32x32 and downsample 64x64 -> 32x32
    for (int p = t; p < 1024; p += 256) {
        const int y = p >> 5, x = p & 31;

        // coarse: src grid 16, scale (H-1)/(oh-1) = 15/31
        float fy = (float)y * (15.0f / 31.0f);
        float fx = (float)x * (15.0f / 31.0f);
        int y0 = (int)floorf(fy), x0 = (int)floorf(fx);
        int y1 = min(y0 + 1, 15), x1 = min(x0 + 1, 15);
        float wy = fy - (float)y0, wx = fx - (float)x0;
        float tp = cf[y0 * 16 + x0] * (1.0f - wx) + cf[y0 * 16 + x1] * wx;
        float bt = cf[y1 * 16 + x0] * (1.0f - wx) + cf[y1 * 16 + x1] * wx;
        fused[0][y + 1][x + 1] = tp * (1.0f - wy) + bt * wy;

        // fine: src grid 64, scale 63/31
        fy = (float)y * (63.0f / 31.0f);
        fx = (float)x * (63.0f / 31.0f);
        y0 = (int)floorf(fy); x0 = (int)floorf(fx);
        y1 = min(y0 + 1, 63); x1 = min(x0 + 1, 63);
        wy = fy - (float)y0;  wx = fx - (float)x0;
        tp = fm[y0 * 64 + x0] * (1.0f - wx) + fm[y0 * 64 + x1] * wx;
        bt = fm[y1 * 64 + x0] * (1.0f - wx) + fm[y1 * 64 + x1] * wx;
        fused[1][y + 1][x + 1] = tp * (1.0f - wy) + bt * wy;
    }
    __syncthreads();

    // conv1: 2->2, 3x3 SAME, ReLU
    for (int p = t; p < 1024; p += 256) {
        const int y = p >> 5, x = p & 31;
#pragma unroll
        for (int oc = 0; oc < 2; ++oc) {
            float acc = bc1[oc];
#pragma unroll
            for (int ic = 0; ic < 2; ++ic)
#pragma unroll
                for (int ky = 0; ky < 3; ++ky)
#pragma unroll
                    for (int kx = 0; kx < 3; ++kx)
                        acc += wc1[((oc * 2 + ic) * 3 + ky) * 3 + kx] *
                               fused[ic][y + ky][x + kx];
            h[oc][y + 1][x + 1] = fmaxf(acc, 0.0f);
        }
    }
    __syncthreads();

    // conv2: 2->1, 3x3 SAME
    float* orow = out + (size_t)b * 1024;
    for (int p = t; p < 1024; p += 256) {
        const int y = p >> 5, x = p & 31;
        float acc = bc2s;
#pragma unroll
        for (int ic = 0; ic < 2; ++ic)
#pragma unroll
            for (int ky = 0; ky < 3; ++ky)
#pragma unroll
                for (int kx = 0; kx < 3; ++kx)
                    acc += wc2[(ic * 3 + ky) * 3 + kx] * h[ic][y + ky][x + kx];
        orow[p] = acc;
    }
}

// ---------------------------------------------------------------------------
extern "C" void kernel_launch(void* const* d_in, const int* in_sizes, int n_in,
                              void* d_out, int out_size, void* d_ws, size_t ws_size,
                              hipStream_t stream)
{
    const float* tokens   = (const float*)d_in[0];
    const float* w_coarse = (const float*)d_in[1];
    const float* b_coarse = (const float*)d_in[2];
    const float* w_fine   = (const float*)d_in[3];
    const float* b_fine   = (const float*)d_in[4];
    const float* w_conv1  = (const float*)d_in[5];
    const float* b_conv1  = (const float*)d_in[6];
    const float* w_conv2  = (const float*)d_in[7];
    const float* b_conv2  = (const float*)d_in[8];
    const int*   mask     = (const int*)d_in[9];

    float* ws_f         = (float*)d_ws;
    float* coarse_feats = ws_f;                                  // B*256
    float* fine_feats   = coarse_feats + (size_t)BB * NC;        // B*4096
    float* fine_map     = fine_feats   + (size_t)BB * NF;        // B*4096

    // K1: 512 batches * 272 tiles = 139264 waves, 8 waves / 256-thread block
    const int tiles_total = BB * (NTOK / 16);
    proj_kernel<<<tiles_total / 8, 256, 0, stream>>>(
        tokens, w_coarse, b_coarse, w_fine, b_fine, coarse_feats, fine_feats);

    scatter_kernel<<<BB, 256, 0, stream>>>(mask, fine_feats, fine_map);

    decode_kernel<<<BB, 256, 0, stream>>>(coarse_feats, fine_map,
                                          w_conv1, b_conv1, w_conv2, b_conv2,
                                          (float*)d_out);
}